// GCNConvLayer_81535659147824
// MI455X (gfx1250) — compile-verified
//
#include <hip/hip_runtime.h>
#include <hip/hip_bf16.h>

typedef __attribute__((ext_vector_type(2))) float v2f;
typedef __attribute__((ext_vector_type(8))) float v8f;

#define DIM 128

// ---------------- degree kernels ----------------
__global__ void deg_init_kernel(float* __restrict__ deg, int N) {
    int i = blockIdx.x * blockDim.x + threadIdx.x;
    if (i < N) deg[i] = 1.0f;  // self loop contributes 1 to every node's degree
}

__global__ void deg_edges_kernel(const long long* __restrict__ ei, float* __restrict__ deg,
                                 int E) {
    int e = blockIdx.x * blockDim.x + threadIdx.x;
    if (e < E) {
        long long col = ei[(long long)E + e];  // target node
        atomicAdd(&deg[col], 1.0f);
    }
}

__global__ void dinv_kernel(const float* __restrict__ deg, float* __restrict__ dinv, int N) {
    int i = blockIdx.x * blockDim.x + threadIdx.x;
    if (i < N) {
        float d = deg[i];
        dinv[i] = (d > 0.0f) ? rsqrtf(d) : 0.0f;
    }
}

// ---------------- GEMM: h = x @ W^T via V_WMMA_F32_16X16X4_F32 ----------------
// Block = 256 threads = 8 waves. Block b handles rows [16b, 16b+16).
// Wave w handles output columns [16w, 16w+16). K = 128 stepped by 4.
//
// 32-bit A 16x4 lane layout (ISA 7.12.2): lane l holds row M = l&15,
//   VGPR0 = K = (l>>4)*2, VGPR1 = K+1.
// 32-bit B 4x16 mirrored: lane l holds col N = l&15, VGPR0/1 = K = (l>>4)*2, K+1.
// Since B[k][n] = W[n][k] and W is row-major (K contiguous), both fragments are
// 8-byte contiguous loads per lane.
__global__ __launch_bounds__(256) void gemm_wmma_kernel(const float* __restrict__ x,
                                                        const float* __restrict__ W,
                                                        float* __restrict__ h, int N) {
    const int lane = threadIdx.x & 31;
    const int wave = threadIdx.x >> 5;          // 0..7 -> N tile
    const int m0   = blockIdx.x * 16;           // row tile base
    const int n0   = wave * 16;                 // col tile base

    const int r15 = lane & 15;
    const int kk  = (lane >> 4) * 2;            // 0 or 2

    const float* __restrict__ xrow = x + (size_t)(m0 + r15) * DIM; // A: row M=r15
    const float* __restrict__ wrow = W + (size_t)(n0 + r15) * DIM; // B: col N=r15 -> W row

    v8f c = {};
#pragma unroll
    for (int k = 0; k < DIM; k += 4) {
        v2f a = *(const v2f*)(xrow + k + kk);
        v2f b = *(const v2f*)(wrow + k + kk);
        c = __builtin_amdgcn_wmma_f32_16x16x4_f32(
            /*neg_a=*/false, a, /*neg_b=*/false, b,
            /*c_mod=*/(short)0, c, /*reuse_a=*/false, /*reuse_b=*/false);
    }

    // C/D layout: VGPR r, lane l -> M = 8*(l>>4) + r, N = l&15
    const int nn    = lane & 15;
    const int mbase = m0 + (lane >> 4) * 8;
#pragma unroll
    for (int r = 0; r < 8; ++r) {
        h[(size_t)(mbase + r) * DIM + n0 + nn] = c[r];
    }
}

// ---------------- out init: self-loop message + bias ----------------
__global__ void out_init_kernel(const float* __restrict__ h, const float* __restrict__ dinv,
                                const float* __restrict__ bias, float* __restrict__ out,
                                long long total) {
    long long idx = (long long)blockIdx.x * blockDim.x + threadIdx.x;
    if (idx < total) {
        int node = (int)(idx >> 7);   // /128
        int d    = (int)(idx & 127);
        float s  = dinv[node];
        out[idx] = s * s * h[idx] + bias[d];
    }
}

// ---------------- edge scatter: one wave32 per edge, float4 per lane ----------------
__global__ __launch_bounds__(256) void edge_scatter_kernel(const long long* __restrict__ ei,
                                                           const float* __restrict__ h,
                                                           const float* __restrict__ dinv,
                                                           float* __restrict__ out, int E) {
    const int lane = threadIdx.x & 31;
    const int e    = (int)((blockIdx.x * blockDim.x + threadIdx.x) >> 5);
    if (e >= E) return;

    const long long r = ei[e];                   // source (gather h[r])
    const long long c = ei[(long long)E + e];    // target (scatter-add out[c])
    const float norm  = dinv[r] * dinv[c];

    const float4 v = ((const float4*)(h + r * DIM))[lane];
    float* dst = out + c * DIM + lane * 4;
    atomicAdd(dst + 0, norm * v.x);
    atomicAdd(dst + 1, norm * v.y);
    atomicAdd(dst + 2, norm * v.z);
    atomicAdd(dst + 3, norm * v.w);
}

extern "C" void kernel_launch(void* const* d_in, const int* in_sizes, int n_in,
                              void* d_out, int out_size, void* d_ws, size_t ws_size,
                              hipStream_t stream) {
    const float*     x    = (const float*)d_in[0];
    const float*     W    = (const float*)d_in[1];
    const float*     bias = (const float*)d_in[2];
    const long long* ei   = (const long long*)d_in[3];

    const int N = in_sizes[0] / DIM;       // 100000
    const int E = in_sizes[3] / 2;         // 1600000
    float* out = (float*)d_out;

    // workspace layout: h[N*128] | deg[N] | dinv[N]
    float* h    = (float*)d_ws;
    float* deg  = h + (size_t)N * DIM;
    float* dinv = deg + N;

    // 1) degree (self loops + edges)
    deg_init_kernel<<<(N + 255) / 256, 256, 0, stream>>>(deg, N);
    deg_edges_kernel<<<(E + 255) / 256, 256, 0, stream>>>(ei, deg, E);
    dinv_kernel<<<(N + 255) / 256, 256, 0, stream>>>(deg, dinv, N);

    // 2) h = x @ W^T  (N/16 row tiles; 8 waves per block cover all 128 cols)
    gemm_wmma_kernel<<<N / 16, 256, 0, stream>>>(x, W, h, N);

    // 3) initialize out with self-loop message + bias
    long long total = (long long)N * DIM;
    out_init_kernel<<<(int)((total + 255) / 256), 256, 0, stream>>>(h, dinv, bias, out, total);

    // 4) scatter-add edge messages (one wave per edge)
    long long threads = (long long)E * 32;
    edge_scatter_kernel<<<(int)((threads + 255) / 256), 256, 0, stream>>>(ei, h, dinv, out, E);
}